// SE_GATTransfomerOnlyDecoderNoEdges_39479339384978
// MI455X (gfx1250) — compile-verified
//
#include <hip/hip_runtime.h>
#include <math.h>

// ---------------------------------------------------------------------------
// CDNA5 (gfx1250) implementation: GAT x4 + transformer decoder layer + pool.
// Dense GEMMs on V_WMMA_F32_16X16X4_F32; weights staged to LDS per workgroup
// via the gfx1250 async global->LDS path (ASYNCcnt) when available.
// ---------------------------------------------------------------------------

typedef float v2f __attribute__((ext_vector_type(2)));
typedef float v8f __attribute__((ext_vector_type(8)));

#define WMMA_F32(a, b, c) \
  __builtin_amdgcn_wmma_f32_16x16x4_f32(false, (a), false, (b), (short)0, (c), false, false)

#if defined(__has_builtin)
#if __has_builtin(__builtin_amdgcn_global_load_async_to_lds_b128)
#define HAVE_ASYNC_LDS 1
#endif
#endif
#ifndef HAVE_ASYNC_LDS
#define HAVE_ASYNC_LDS 0
#endif

#if HAVE_ASYNC_LDS
// exact parameter types per hipcc diagnostic: AS1 / AS3 pointers to int __vector(4)
typedef int vsi4 __attribute__((__vector_size__(4 * sizeof(int))));
typedef __attribute__((address_space(1))) vsi4* gv4p;
typedef __attribute__((address_space(3))) vsi4* lv4p;
#endif

constexpr int N_GRAPHS = 128;
constexpr int N_NODES  = 512;
constexpr int NTOT     = N_GRAPHS * N_NODES;   // 65536
constexpr int E_EDGES  = 524288;
constexpr int EE       = E_EDGES + NTOT;       // edges + self loops
constexpr int HEADS    = 4;                    // GAT heads
constexpr int GDIM     = 16;
constexpr int DMODEL   = HEADS * GDIM;         // 64
constexpr int NHEAD    = 2;                    // attention heads
constexpr int HD       = DMODEL / NHEAD;       // 32

// ---------------------------------------------------------------------------
// Wave-level GEMM: 16-row tile x 64 cols, K = KDIM, f32 WMMA.
// A: [nrows, KDIM] row-major, W: [KDIM, 64] row-major.
// W is staged into LDS once per workgroup (async global->LDS when available),
// B-fragments are then served from LDS by all 8 waves.
// EPI: 0 = none, 1 = +bias, 2 = +bias then ReLU.
// A-frag layout (16x4 f32): lane L holds A[L&15][kbase + ((L>>4)<<1) + {0,1}]
// B-frag layout (4x16 f32): lane L holds B[kbase + ((L>>4)<<1) + {0,1}][L&15]
// C/D layout: VGPR r: lanes0-15 -> M=r, lanes16-31 -> M=8+r, N = L&15
// ---------------------------------------------------------------------------
template <int KDIM, int EPI>
__global__ void gemm_n64_kernel(const float* __restrict__ A, const float* __restrict__ W,
                                const float* __restrict__ bias, float* __restrict__ Out,
                                int nrows) {
  __shared__ float sW[KDIM * 64];

  // ---- stage W ([KDIM,64] f32) into LDS, one copy per workgroup ----
#if HAVE_ASYNC_LDS
  {
    constexpr int CHUNKS = KDIM * 16;  // 16-byte chunks
    for (int c = threadIdx.x; c < CHUNKS; c += blockDim.x) {
      __builtin_amdgcn_global_load_async_to_lds_b128((gv4p)(W + c * 4), (lv4p)(&sW[c * 4]), 0, 0);
    }
#if defined(__has_builtin) && __has_builtin(__builtin_amdgcn_s_wait_asynccnt)
    __builtin_amdgcn_s_wait_asynccnt(0);
#else
    asm volatile("s_wait_asynccnt 0" ::: "memory");
#endif
  }
#else
  for (int c = threadIdx.x; c < KDIM * 16; c += blockDim.x) {
    const float4 v = ((const float4*)W)[c];
    ((float4*)sW)[c] = v;
  }
#endif
  __syncthreads();

  const int lane = threadIdx.x & 31;
  const int wave = threadIdx.x >> 5;
  const int rowBase = (blockIdx.x * (blockDim.x >> 5) + wave) * 16;
  if (rowBase >= nrows) return;
  const float* Ar = A + (size_t)rowBase * KDIM;
  const int m = lane & 15;
  const int koff = (lane >> 4) << 1;
  const int n = lane & 15;
  for (int ct = 0; ct < 4; ++ct) {
    v8f c = {};
#pragma unroll
    for (int ks = 0; ks < KDIM / 4; ++ks) {
      const int k0 = ks * 4 + koff;
      v2f a;
      a.x = Ar[m * KDIM + k0];
      a.y = Ar[m * KDIM + k0 + 1];
      v2f b;
      b.x = sW[k0 * 64 + ct * 16 + n];
      b.y = sW[(k0 + 1) * 64 + ct * 16 + n];
      c = WMMA_F32(a, b, c);
    }
    const int col = ct * 16 + n;
    float bv = 0.f;
    if (EPI >= 1) bv = bias[col];
#pragma unroll
    for (int r = 0; r < 8; ++r) {
      const int row = rowBase + ((lane >> 4) << 3) + r;
      float v = c[r] + bv;
      if (EPI == 2) v = fmaxf(v, 0.f);
      Out[(size_t)row * 64 + col] = v;
    }
  }
}

// ---------------------------------------------------------------------------
// feature_fc: h = x @ Wf + bf   ([N,4] @ [4,4])
// ---------------------------------------------------------------------------
__global__ void featfc_kernel(const float* __restrict__ x, const float* __restrict__ Wf,
                              const float* __restrict__ bf, float* __restrict__ h) {
  const int nidx = blockIdx.x * blockDim.x + threadIdx.x;
  if (nidx >= NTOT) return;
  float xv[4];
#pragma unroll
  for (int k = 0; k < 4; ++k) xv[k] = x[nidx * 4 + k];
#pragma unroll
  for (int j = 0; j < 4; ++j) {
    float a = bf[j];
#pragma unroll
    for (int k = 0; k < 4; ++k) a += xv[k] * Wf[k * 4 + j];
    h[nidx * 4 + j] = a;
  }
}

// ---------------------------------------------------------------------------
// GAT edge machinery
// ---------------------------------------------------------------------------
__global__ void esed_kernel(const float* __restrict__ hW, const float* __restrict__ aS,
                            const float* __restrict__ aD, float* __restrict__ es,
                            float* __restrict__ ed) {
  const int idx = blockIdx.x * blockDim.x + threadIdx.x;  // n*HEADS + h
  if (idx >= NTOT * HEADS) return;
  const int nidx = idx >> 2, h = idx & 3;
  const float* hp = hW + (size_t)nidx * 64 + h * 16;
  const float* as = aS + h * 16;
  const float* ad = aD + h * 16;
  float s1 = 0.f, s2 = 0.f;
#pragma unroll
  for (int o = 0; o < 16; ++o) {
    const float v = hp[o];
    s1 += v * as[o];
    s2 += v * ad[o];
  }
  es[idx] = s1;
  ed[idx] = s2;
}

__global__ void gat_init_kernel(float* __restrict__ m, float* __restrict__ ssum,
                                float* __restrict__ agg) {
  const int i = blockIdx.x * blockDim.x + threadIdx.x;
  if (i < NTOT * HEADS) {
    m[i] = -1e30f;
    ssum[i] = 0.f;
  }
  if (i < NTOT * 64) agg[i] = 0.f;
}

__device__ __forceinline__ void edge_sd(int e, const int* __restrict__ ei, int& s, int& d) {
  if (e < E_EDGES) {
    s = ei[e];
    d = ei[E_EDGES + e];
  } else {
    s = e - E_EDGES;
    d = s;
  }
}

__device__ __forceinline__ void atomicMaxF(float* addr, float val) {
  int* ai = (int*)addr;
  int old = __float_as_int(*addr);
  while (__int_as_float(old) < val) {
    const int assumed = old;
    old = atomicCAS(ai, assumed, __float_as_int(val));
    if (old == assumed) break;
  }
}

__global__ void edge_max_kernel(const int* __restrict__ ei, const float* __restrict__ es,
                                const float* __restrict__ ed, float* __restrict__ ebuf,
                                float* __restrict__ m) {
  const int idx = blockIdx.x * blockDim.x + threadIdx.x;  // e*HEADS + h
  if (idx >= EE * HEADS) return;
  const int e = idx >> 2, h = idx & 3;
  int s, d;
  edge_sd(e, ei, s, d);
  float v = es[s * 4 + h] + ed[d * 4 + h];
  v = v > 0.f ? v : 0.2f * v;  // leaky_relu 0.2
  ebuf[idx] = v;
  atomicMaxF(&m[d * 4 + h], v);
}

__global__ void edge_sum_kernel(const int* __restrict__ ei, const float* __restrict__ m,
                                float* __restrict__ ebuf, float* __restrict__ ssum) {
  const int idx = blockIdx.x * blockDim.x + threadIdx.x;
  if (idx >= EE * HEADS) return;
  const int e = idx >> 2, h = idx & 3;
  int s, d;
  edge_sd(e, ei, s, d);
  const float w = __expf(ebuf[idx] - m[d * 4 + h]);
  ebuf[idx] = w;
  atomicAdd(&ssum[d * 4 + h], w);
}

__global__ void edge_agg_kernel(const int* __restrict__ ei, const float* __restrict__ ebuf,
                                const float* __restrict__ ssum, const float* __restrict__ hW,
                                float* __restrict__ agg) {
  const int idx = blockIdx.x * blockDim.x + threadIdx.x;
  if (idx >= EE * HEADS) return;
  const int e = idx >> 2, h = idx & 3;
  int s, d;
  edge_sd(e, ei, s, d);
  const float alpha = ebuf[idx] / (ssum[d * 4 + h] + 1e-16f);
  const float* hp = hW + (size_t)s * 64 + h * 16;
  float* op = agg + (size_t)d * 64 + h * 16;
#pragma unroll
  for (int o = 0; o < 16; ++o) atomicAdd(&op[o], alpha * hp[o]);
}

__global__ void gat_final_kernel(const float* __restrict__ agg, const float* __restrict__ gb,
                                 float* __restrict__ hout) {
  const int i = blockIdx.x * blockDim.x + threadIdx.x;
  if (i >= NTOT * 64) return;
  const float v = agg[i] + gb[i & 63];
  hout[i] = v > 0.f ? v : 0.01f * v;  // leaky_relu 0.01
}

// ---------------------------------------------------------------------------
// Flash-style attention: one wave per (graph, head, 16-query tile).
// Scores tile 16x512 lives in LDS (33 KB << 320 KB/WGP). QK^T and P*V on WMMA.
// ---------------------------------------------------------------------------
__global__ void attn_kernel(const float* __restrict__ Q, const float* __restrict__ K,
                            const float* __restrict__ V, float* __restrict__ O) {
  __shared__ float sc[16][520];  // padded to dodge bank conflicts
  const int lane = threadIdx.x;  // blockDim.x == 32
  const int qt = blockIdx.x & 31;
  const int head = (blockIdx.x >> 5) & (NHEAD - 1);
  const int g = blockIdx.x >> 6;
  const float* Qp = Q + (size_t)g * N_NODES * 64 + head * HD;
  const float* Kp = K + (size_t)g * N_NODES * 64 + head * HD;
  const float* Vp = V + (size_t)g * N_NODES * 64 + head * HD;
  float* Op = O + (size_t)g * N_NODES * 64 + head * HD;
  const int m = lane & 15;
  const int koff = (lane >> 4) << 1;
  const int n = lane & 15;
  const int qbase = qt * 16;
  const float scale = 0.1767766952966369f;  // 1/sqrt(32)

  // scores = Q K^T / sqrt(hd)
  for (int kt = 0; kt < N_NODES / 16; ++kt) {
    v8f c = {};
#pragma unroll
    for (int ks = 0; ks < HD / 4; ++ks) {
      const int k0 = ks * 4 + koff;
      v2f a;
      a.x = Qp[(size_t)(qbase + m) * 64 + k0];
      a.y = Qp[(size_t)(qbase + m) * 64 + k0 + 1];
      v2f b;  // B[k][n] = K[kt*16+n][k]  (transposed access)
      b.x = Kp[(size_t)(kt * 16 + n) * 64 + k0];
      b.y = Kp[(size_t)(kt * 16 + n) * 64 + k0 + 1];
      c = WMMA_F32(a, b, c);
    }
#pragma unroll
    for (int r = 0; r < 8; ++r) sc[((lane >> 4) << 3) + r][kt * 16 + n] = c[r] * scale;
  }
  __syncthreads();

  // row softmax (16 rows, one lane each)
  if (lane < 16) {
    float mx = -1e30f;
    for (int j = 0; j < N_NODES; ++j) mx = fmaxf(mx, sc[lane][j]);
    float sum = 0.f;
    for (int j = 0; j < N_NODES; ++j) {
      const float w = __expf(sc[lane][j] - mx);
      sc[lane][j] = w;
      sum += w;
    }
    const float inv = 1.f / sum;
    for (int j = 0; j < N_NODES; ++j) sc[lane][j] *= inv;
  }
  __syncthreads();

  // O = P @ V : contraction over 512 keys
  for (int ct = 0; ct < 2; ++ct) {
    v8f c = {};
    for (int ks = 0; ks < N_NODES / 4; ++ks) {
      const int k0 = ks * 4 + koff;
      v2f a;
      a.x = sc[m][k0];
      a.y = sc[m][k0 + 1];
      v2f b;
      b.x = Vp[(size_t)k0 * 64 + ct * 16 + n];
      b.y = Vp[(size_t)(k0 + 1) * 64 + ct * 16 + n];
      c = WMMA_F32(a, b, c);
    }
#pragma unroll
    for (int r = 0; r < 8; ++r)
      Op[(size_t)(qbase + ((lane >> 4) << 3) + r) * 64 + ct * 16 + n] = c[r];
  }
}

// ---------------------------------------------------------------------------
// LayerNorm(A + B) over last dim (64). One wave per row, 2 cols/lane, shfl_xor.
// ---------------------------------------------------------------------------
__global__ void ln_add_kernel(const float* __restrict__ A, const float* __restrict__ B,
                              const float* __restrict__ g, const float* __restrict__ b,
                              float* __restrict__ Out) {
  const int lane = threadIdx.x & 31;
  const int wave = threadIdx.x >> 5;
  const int row = blockIdx.x * (blockDim.x >> 5) + wave;
  if (row >= NTOT) return;
  const size_t base = (size_t)row * 64;
  const float x0 = A[base + lane] + B[base + lane];
  const float x1 = A[base + 32 + lane] + B[base + 32 + lane];
  float s = x0 + x1;
#pragma unroll
  for (int off = 16; off >= 1; off >>= 1) s += __shfl_xor(s, off, 32);
  const float mean = s * (1.f / 64.f);
  const float d0 = x0 - mean, d1 = x1 - mean;
  float v = d0 * d0 + d1 * d1;
#pragma unroll
  for (int off = 16; off >= 1; off >>= 1) v += __shfl_xor(v, off, 32);
  const float inv = rsqrtf(v * (1.f / 64.f) + 1e-5f);
  Out[base + lane] = d0 * inv * g[lane] + b[lane];
  Out[base + 32 + lane] = d1 * inv * g[32 + lane] + b[32 + lane];
}

// ---------------------------------------------------------------------------
// Mean pool per graph
// ---------------------------------------------------------------------------
__global__ void pool_kernel(const float* __restrict__ t, float* __restrict__ pooled) {
  const int idx = blockIdx.x * blockDim.x + threadIdx.x;  // g*64 + d
  if (idx >= N_GRAPHS * 64) return;
  const int g = idx >> 6, d = idx & 63;
  const float* p = t + (size_t)g * N_NODES * 64 + d;
  float s = 0.f;
  for (int j = 0; j < N_NODES; ++j) s += p[(size_t)j * 64];
  pooled[idx] = s * (1.f / N_NODES);
}

// ---------------------------------------------------------------------------
// Host launch
// ---------------------------------------------------------------------------
extern "C" void kernel_launch(void* const* d_in, const int* in_sizes, int n_in,
                              void* d_out, int out_size, void* d_ws, size_t ws_size,
                              hipStream_t stream) {
  (void)in_sizes; (void)n_in; (void)out_size; (void)ws_size;
  const float* x   = (const float*)d_in[0];
  const int*   ei  = (const int*)d_in[1];
  // d_in[2] = batch (structure is known: repeat(arange(128), 512)) -> unused
  const float* Wf  = (const float*)d_in[3];
  const float* bf  = (const float*)d_in[4];
  const float* gW0 = (const float*)d_in[5];
  const float* aS0 = (const float*)d_in[6];
  const float* aD0 = (const float*)d_in[7];
  const float* gb0 = (const float*)d_in[8];
  const float* gW  = (const float*)d_in[9];
  const float* aS  = (const float*)d_in[10];
  const float* aD  = (const float*)d_in[11];
  const float* gb  = (const float*)d_in[12];
  const float* sWq = (const float*)d_in[13];
  const float* sWk = (const float*)d_in[14];
  const float* sWv = (const float*)d_in[15];
  const float* sWo = (const float*)d_in[16];
  const float* cWq = (const float*)d_in[17];
  const float* cWk = (const float*)d_in[18];
  const float* cWv = (const float*)d_in[19];
  const float* cWo = (const float*)d_in[20];
  const float* sbq = (const float*)d_in[21];
  const float* sbk = (const float*)d_in[22];
  const float* sbv = (const float*)d_in[23];
  const float* sbo = (const float*)d_in[24];
  const float* cbq = (const float*)d_in[25];
  const float* cbk = (const float*)d_in[26];
  const float* cbv = (const float*)d_in[27];
  const float* cbo = (const float*)d_in[28];
  const float* ln1g = (const float*)d_in[29];
  const float* ln1b = (const float*)d_in[30];
  const float* ln2g = (const float*)d_in[31];
  const float* ln2b = (const float*)d_in[32];
  const float* ln3g = (const float*)d_in[33];
  const float* ln3b = (const float*)d_in[34];
  const float* ffW1 = (const float*)d_in[35];
  const float* ffb1 = (const float*)d_in[36];
  const float* ffW2 = (const float*)d_in[37];
  const float* ffb2 = (const float*)d_in[38];
  const float* fcW  = (const float*)d_in[39];
  const float* fcb  = (const float*)d_in[40];
  float* out = (float*)d_out;

  // workspace layout: 6 slabs of [NTOT,64] f32 + small region
  const size_t SLAB = (size_t)NTOT * 64 * sizeof(float);  // 16 MiB
  float* slab[6];
  for (int i = 0; i < 6; ++i) slab[i] = (float*)((char*)d_ws + (size_t)i * SLAB);
  char* sm = (char*)d_ws + 6 * SLAB;
  float* es_buf  = (float*)sm;                 sm += (size_t)NTOT * 4 * sizeof(float);
  float* ed_buf  = (float*)sm;                 sm += (size_t)NTOT * 4 * sizeof(float);
  float* m_buf   = (float*)sm;                 sm += (size_t)NTOT * 4 * sizeof(float);
  float* s_buf   = (float*)sm;                 sm += (size_t)NTOT * 4 * sizeof(float);
  float* h0_buf  = (float*)sm;                 sm += (size_t)NTOT * 4 * sizeof(float);
  float* pooled  = (float*)sm;                 sm += (size_t)N_GRAPHS * 64 * sizeof(float);

  const dim3 B256(256);
  const int gN      = (NTOT + 255) / 256;            // per-node threads
  const int gNH     = (NTOT * 4 + 255) / 256;        // per (node,head)
  const int gNE     = (NTOT * 64 + 255) / 256;       // per (node,feature)
  const int gEH     = (EE * 4 + 255) / 256;          // per (edge,head)
  const int gGemmN  = NTOT / (16 * 8);               // 16-row tiles, 8 waves/block
  const int gLN     = NTOT / 8;                      // 8 rows/block

  // -------- feature fc --------
  featfc_kernel<<<gN, B256, 0, stream>>>(x, Wf, bf, h0_buf);

  // -------- GAT layers (ping-pong slab0 <-> slab1; hW in slab2, ebuf in slab3) --------
  float* hW   = slab[2];
  float* ebuf = slab[3];
  auto gat_edges = [&](const float* aSl, const float* aDl, const float* gbl, float* aggout) {
    esed_kernel<<<gNH, B256, 0, stream>>>(hW, aSl, aDl, es_buf, ed_buf);
    gat_init_kernel<<<gNE, B256, 0, stream>>>(m_buf, s_buf, aggout);
    edge_max_kernel<<<gEH, B256, 0, stream>>>(ei, es_buf, ed_buf, ebuf, m_buf);
    edge_sum_kernel<<<gEH, B256, 0, stream>>>(ei, m_buf, ebuf, s_buf);
    edge_agg_kernel<<<gEH, B256, 0, stream>>>(ei, ebuf, s_buf, hW, aggout);
    gat_final_kernel<<<gNE, B256, 0, stream>>>(aggout, gbl, aggout);  // in-place ok
  };

  // layer 0: [N,4] -> [N,64], output -> slab1
  gemm_n64_kernel<4, 0><<<gGemmN, B256, 0, stream>>>(h0_buf, gW0, nullptr, hW, NTOT);
  gat_edges(aS0, aD0, gb0, slab[1]);
  // layers 1..3: ping-pong, final lands in slab0
  float* hin = slab[1];
  for (int i = 0; i < 3; ++i) {
    float* hout = (hin == slab[1]) ? slab[0] : slab[1];
    gemm_n64_kernel<64, 0><<<gGemmN, B256, 0, stream>>>(hin, gW + (size_t)i * 64 * 64,
                                                        nullptr, hW, NTOT);
    gat_edges(aS + (size_t)i * HEADS * GDIM, aD + (size_t)i * HEADS * GDIM,
              gb + (size_t)i * 64, hout);
    hin = hout;
  }
  float* t = slab[0];  // decoder input; also cross-attn memory

  const int gAttn = N_GRAPHS * NHEAD * (N_NODES / 16);  // 8192 waves

  // -------- self-attention --------
  gemm_n64_kernel<64, 1><<<gGemmN, B256, 0, stream>>>(t, sWq, sbq, slab[1], NTOT);
  gemm_n64_kernel<64, 1><<<gGemmN, B256, 0, stream>>>(t, sWk, sbk, slab[2], NTOT);
  gemm_n64_kernel<64, 1><<<gGemmN, B256, 0, stream>>>(t, sWv, sbv, slab[3], NTOT);
  attn_kernel<<<gAttn, dim3(32), 0, stream>>>(slab[1], slab[2], slab[3], slab[4]);
  gemm_n64_kernel<64, 1><<<gGemmN, B256, 0, stream>>>(slab[4], sWo, sbo, slab[5], NTOT);
  ln_add_kernel<<<gLN, B256, 0, stream>>>(t, slab[5], ln1g, ln1b, slab[1]);  // t1 = slab1

  // -------- cross-attention (memory = t) --------
  gemm_n64_kernel<64, 1><<<gGemmN, B256, 0, stream>>>(slab[1], cWq, cbq, slab[2], NTOT);
  gemm_n64_kernel<64, 1><<<gGemmN, B256, 0, stream>>>(t, cWk, cbk, slab[3], NTOT);
  gemm_n64_kernel<64, 1><<<gGemmN, B256, 0, stream>>>(t, cWv, cbv, slab[4], NTOT);
  attn_kernel<<<gAttn, dim3(32), 0, stream>>>(slab[2], slab[3], slab[4], slab[5]);
  gemm_n64_kernel<64, 1><<<gGemmN, B256, 0, stream>>>(slab[5], cWo, cbo, slab[2], NTOT);
  ln_add_kernel<<<gLN, B256, 0, stream>>>(slab[1], slab[2], ln2g, ln2b, slab[3]);  // t2 = slab3

  // -------- FFN --------
  gemm_n64_kernel<64, 2><<<gGemmN, B256, 0, stream>>>(slab[3], ffW1, ffb1, slab[4], NTOT);
  gemm_n64_kernel<64, 1><<<gGemmN, B256, 0, stream>>>(slab[4], ffW2, ffb2, slab[5], NTOT);
  ln_add_kernel<<<gLN, B256, 0, stream>>>(slab[3], slab[5], ln3g, ln3b, slab[0]);  // t3 = slab0

  // -------- pool + final fc --------
  pool_kernel<<<(N_GRAPHS * 64 + 255) / 256, B256, 0, stream>>>(slab[0], pooled);
  gemm_n64_kernel<64, 1><<<1, B256, 0, stream>>>(pooled, fcW, fcb, out, N_GRAPHS);
}